// SpatialProductLayer_16088947491590
// MI455X (gfx1250) — compile-verified
//
#include <hip/hip_runtime.h>

typedef __attribute__((ext_vector_type(2))) float v2f;
typedef __attribute__((ext_vector_type(8))) float v8f;

#define IN_C 32
#define IN_H 32
#define IN_W 32
#define OUT_C 64
#define OH 33
#define OW 33
#define NPOS (OH * OW)              // 1089
#define PADW 34                     // padded cols: index = iw+1, range 0..33
#define PADH 35                     // rows 0..33 real (ih+1), row 34 dead (tail reads)
#define CH_STRIDE (PADH * PADW)     // 1190 floats per channel
#define SXN (IN_C * CH_STRIDE)      // 38080 floats = 148.75 KB
#define THREADS 1024
#define XBATCH (IN_C * IN_H * IN_W) // 32768 floats

#if __has_builtin(__builtin_amdgcn_global_load_async_to_lds_b32) && \
    __has_builtin(__builtin_amdgcn_s_wait_asynccnt)
#define USE_ASYNC_LDS 1
#endif

__global__ __launch_bounds__(THREADS)
void spatial_product_kernel(const float* __restrict__ x,
                            const float* __restrict__ w,
                            float* __restrict__ out)
{
  __shared__ float s_x[SXN];          // zero-padded input image for this batch
  __shared__ int   s_idx[OUT_C * 4];  // one-hot channel index per (oc, tap)

  const int tid = threadIdx.x;
  const int b   = blockIdx.x;

  // Recover idx[oc][k] = argmax_ic weight[oc, ic, kh, kw] from the one-hot weight.
  if (tid < OUT_C * 4) {
    const int oc = tid >> 2, k = tid & 3;
    const int kh = k >> 1, kw = k & 1;
    int idx = 0;
    for (int ic = 0; ic < IN_C; ++ic)
      if (w[((oc * IN_C + ic) * 2 + kh) * 2 + kw] != 0.0f) idx = ic;
    s_idx[tid] = idx;
  }

  // Zero-fill the whole padded image (borders, dead row, interior).
  {
    float4 z = {0.f, 0.f, 0.f, 0.f};
    float4* dst = (float4*)s_x;
#pragma unroll
    for (int j = 0; j < (SXN / 4 + THREADS - 1) / THREADS; ++j) {
      const int c = j * THREADS + tid;
      if (c < SXN / 4) dst[c] = z;
    }
  }
  __syncthreads();   // all zero-stores land before any async write

  // Stage the interior via the async-to-LDS engine (scatter into padded layout).
  const float* xb = x + (size_t)b * XBATCH;
#ifdef USE_ASYNC_LDS
  {
    typedef __attribute__((address_space(1))) int gi;
    typedef __attribute__((address_space(3))) int li;
    gi* gp = (gi*)xb;
    li* lp = (li*)s_x;
#pragma unroll
    for (int j = 0; j < XBATCH / THREADS; ++j) {
      const int e  = j * THREADS + tid;
      const int ic = e >> 10, ih = (e >> 5) & 31, iw = e & 31;
      const int off = ic * CH_STRIDE + (ih + 1) * PADW + (iw + 1);
      __builtin_amdgcn_global_load_async_to_lds_b32(gp + e, lp + off, 0, 0);
    }
    __builtin_amdgcn_s_wait_asynccnt(0);
  }
#else
  {
#pragma unroll
    for (int j = 0; j < XBATCH / THREADS; ++j) {
      const int e  = j * THREADS + tid;
      const int ic = e >> 10, ih = (e >> 5) & 31, iw = e & 31;
      s_x[ic * CH_STRIDE + (ih + 1) * PADW + (iw + 1)] = xb[e];
    }
  }
#endif
  __syncthreads();

  const int wave = tid >> 5;
  const int lane = tid & 31;
  const int n = lane & 15;   // output column inside the WMMA tile
  const int g = lane >> 4;   // 0 -> taps (kh=0,kw=0/1), 1 -> taps (kh=1,kw=0/1)

  const v2f a_ones = {1.0f, 1.0f};   // A = ones(16x4): WMMA computes per-column tap sums

#pragma unroll
  for (int t = 0; t < 2; ++t) {
    const int oc    = wave * 2 + t;                         // wave-uniform
    const int base0 = s_idx[oc * 4 + g * 2 + 0] * CH_STRIDE; // tap kw=0 channel
    const int base1 = s_idx[oc * 4 + g * 2 + 1] * CH_STRIDE; // tap kw=1 channel
    float* outp = out + ((size_t)b * OUT_C + oc) * NPOS;     // wave-uniform base

    // rowterm = (oh + g)*PADW + ow  for pos = chunk*16 + n, kept incrementally.
    int rt  = g * PADW + n;
    int ow  = n;
    int pos = n;

#pragma unroll 4
    for (int chunk = 0; chunk < 68; ++chunk) {   // chunks 0..67 fully valid
      const float t0 = s_x[base0 + rt];          // tap kw=0: padded col = ow
      const float t1 = s_x[base1 + rt + 1];      // tap kw=1: padded col = ow+1
      v2f bv; bv.x = t0; bv.y = t1;
      v8f c = {};
      v8f d = __builtin_amdgcn_wmma_f32_16x16x4_f32(
          false, a_ones, false, bv, (short)0, c, false, false);
      outp[pos] = d[0];                          // both lane halves write the
                                                 // identical value -> no exec mask
      pos += 16;
      ow += 16;
      const bool wr = ow >= OW;
      ow = wr ? ow - OW : ow;
      rt += wr ? 17 : 16;                        // +16, +PADW-OW on row wrap
    }

    // Tail chunk 68: only column n==0 (pos 1088) is a real output.
    {
      const float t0 = s_x[base0 + rt];          // dead-row/zero reads are safe
      const float t1 = s_x[base1 + rt + 1];
      v2f bv; bv.x = t0; bv.y = t1;
      v8f c = {};
      v8f d = __builtin_amdgcn_wmma_f32_16x16x4_f32(
          false, a_ones, false, bv, (short)0, c, false, false);
      if (n == 0) outp[NPOS - 1] = d[0];
    }
  }
}

extern "C" void kernel_launch(void* const* d_in, const int* in_sizes, int n_in,
                              void* d_out, int out_size, void* d_ws, size_t ws_size,
                              hipStream_t stream) {
  const float* x = (const float*)d_in[0];
  const float* w = (const float*)d_in[1];
  float* out = (float*)d_out;
  const int batch = in_sizes[0] / XBATCH;    // 512
  (void)n_in; (void)out_size; (void)d_ws; (void)ws_size;
  spatial_product_kernel<<<dim3(batch), dim3(THREADS), 0, stream>>>(x, w, out);
}